// SuperfeatureTripletLoss_55353538510995
// MI455X (gfx1250) — compile-verified
//
#include <hip/hip_runtime.h>
#include <hip/hip_bf16.h>
#include <stdint.h>

typedef __attribute__((ext_vector_type(2))) float v2f;
typedef __attribute__((ext_vector_type(8))) float v8f;

#define NPTS 4096
#define DIM  1024
#define NNEG 6
#define LDS_STRIDE 1028   // 1024 + 4 pad floats -> bank-conflict-free frag reads

__device__ __forceinline__ unsigned int okey(float f) {
    unsigned int b = __float_as_uint(f);
    return (b & 0x80000000u) ? ~b : (b | 0x80000000u);
}

// async copy of 256 contiguous bytes-per-thread pattern: 16B per call
__device__ __forceinline__ void async_copy_b128(float* dst_lds, const float* src_glb) {
    unsigned int la = (unsigned int)(unsigned long long)dst_lds;   // low 32 bits = LDS offset
    unsigned long long ga = (unsigned long long)src_glb;
    asm volatile("global_load_async_to_lds_b128 %0, %1, off" :: "v"(la), "v"(ga) : "memory");
}
__device__ __forceinline__ void async_wait0() {
    asm volatile("s_wait_asynccnt 0" ::: "memory");
}

// ---------------------------------------------------------------- invnorm
__global__ __launch_bounds__(256)
void k_invnorm(const float* __restrict__ sf, float* __restrict__ invn) {
    int row  = blockIdx.x * 8 + (threadIdx.x >> 5);
    int lane = threadIdx.x & 31;
    const float* src = sf + (size_t)row * DIM;
    float s = 0.f;
#pragma unroll
    for (int i = 0; i < 8; ++i) {
        float4 v = ((const float4*)src)[lane + i * 32];
        s += v.x*v.x + v.y*v.y + v.z*v.z + v.w*v.w;
    }
#pragma unroll
    for (int m = 16; m >= 1; m >>= 1) s += __shfl_xor(s, m, 32);
    if (lane == 0) invn[row] = 1.0f / fmaxf(sqrtf(s), 1e-12f);
}

// ---------------------------------------------------------------- init u64
__global__ void k_init(unsigned long long* __restrict__ p, int n) {
    int i = blockIdx.x * blockDim.x + threadIdx.x;
    if (i < n) p[i] = 0ull;
}

// ---------------------------------------------------------------- attention top-half mask (rank counting)
__global__ __launch_bounds__(256)
void k_mask(const float* __restrict__ att1, int* __restrict__ cnt) {
    __shared__ float sh[NPTS];
    int tid = threadIdx.x;
    for (int i = tid; i < NPTS; i += 256) sh[i] = att1[i];
    __syncthreads();
    int j = blockIdx.x * 256 + tid;
    float a = sh[j];
    int c = 0;
    for (int k = 0; k < NPTS; ++k) c += (sh[k] > a) ? 1 : 0;
    cnt[j] = c;
}

// ---------------------------------------------------------------- main GEMM: S = qn * pn^T (scales applied in epilogue)
// grid (8 colblocks of 512, 32 rowgroups of 128), 256 threads (8 waves)
__global__ __launch_bounds__(256)
void k_gemm_qp(const float* __restrict__ sf, const float* __restrict__ invn,
               unsigned long long* __restrict__ rowbest, float4* __restrict__ colpart) {
    __shared__ float As[16 * LDS_STRIDE];
    const int cb   = blockIdx.x;
    const int rg   = blockIdx.y;
    const int tid  = threadIdx.x;
    const int wave = tid >> 5;
    const int lane = tid & 31;
    const int half = lane >> 4;
    const int l15  = lane & 15;

    const float* Q = sf;
    const float* P = sf + (size_t)NPTS * DIM;

    const int wcol0 = cb * 512 + wave * 64;
    float sb[4]; const float* pb[4]; int cols[4];
#pragma unroll
    for (int s = 0; s < 4; ++s) {
        int c   = wcol0 + s * 16 + l15;
        cols[s] = c;
        sb[s]   = invn[NPTS + c];
        pb[s]   = P + (size_t)c * DIM;
    }

    float cs1[4], cs2[4]; int ci1[4];
#pragma unroll
    for (int s = 0; s < 4; ++s) { cs1[s] = -3.0e38f; cs2[s] = -3.0e38f; ci1[s] = 0; }

    for (int rt = 0; rt < 8; ++rt) {
        const int row0 = rg * 128 + rt * 16;
        {   // stage raw A tile (16 x 1024) into LDS via async global->LDS
            int r  = tid >> 4;
            int c0 = (tid & 15) * 64;
            const float* src = Q + (size_t)(row0 + r) * DIM + c0;
            float* dst = As + r * LDS_STRIDE + c0;
#pragma unroll
            for (int i = 0; i < 16; ++i)
                async_copy_b128(dst + i * 4, src + i * 4);
            async_wait0();
        }
        __syncthreads();

        v8f acc[4];
#pragma unroll
        for (int s = 0; s < 4; ++s)
#pragma unroll
            for (int r = 0; r < 8; ++r) acc[s][r] = 0.f;

        const float* arow = As + l15 * LDS_STRIDE + half * 2;
#pragma unroll 4
        for (int k0 = 0; k0 < DIM; k0 += 4) {
            v2f aF = *(const v2f*)(arow + k0);
#pragma unroll
            for (int s = 0; s < 4; ++s) {
                v2f bF = *(const v2f*)(pb[s] + k0 + half * 2);
                acc[s] = __builtin_amdgcn_wmma_f32_16x16x4_f32(
                    false, aF, false, bF, (short)0, acc[s], false, false);
            }
        }
        __syncthreads();   // all waves done reading As before restage

        // ---- apply normalization scales in epilogue ----
        float rsc[8];
#pragma unroll
        for (int r = 0; r < 8; ++r) rsc[r] = invn[row0 + r + (half << 3)];
#pragma unroll
        for (int s = 0; s < 4; ++s)
#pragma unroll
            for (int r = 0; r < 8; ++r) acc[s][r] *= rsc[r] * sb[s];

        // ---- per-column top-2 (value+row of best) ----
#pragma unroll
        for (int s = 0; s < 4; ++s) {
            float v1 = -3.0e38f, v2 = -3.0e38f; int r1 = 0;
#pragma unroll
            for (int r = 0; r < 8; ++r) {
                float v = acc[s][r];
                int row = row0 + r + (half << 3);
                if (v > v1) { v2 = v1; v1 = v; r1 = row; }
                else if (v > v2) v2 = v;
            }
            float o1 = __shfl_xor(v1, 16, 32);
            float o2 = __shfl_xor(v2, 16, 32);
            int  orr = __shfl_xor(r1, 16, 32);
            if (half == 0) {
                if (o1 > v1) { v2 = fmaxf(v1, o2); v1 = o1; r1 = orr; }
                else         { v2 = fmaxf(v2, o1); }
            } else {
                if (o1 >= v1) { v2 = fmaxf(v1, o2); v1 = o1; r1 = orr; }
                else          { v2 = fmaxf(v2, o1); }
            }
            if (v1 > cs1[s]) { cs2[s] = fmaxf(cs1[s], v2); cs1[s] = v1; ci1[s] = r1; }
            else             { cs2[s] = fmaxf(cs2[s], v1); }
        }

        // ---- per-row argmax over this wave's 64 cols -> global atomic ----
        float rv[8]; int rc[8];
#pragma unroll
        for (int r = 0; r < 8; ++r) {
            float v = acc[0][r]; int c = cols[0];
#pragma unroll
            for (int s = 1; s < 4; ++s) {
                float w = acc[s][r];
                if (w > v) { v = w; c = cols[s]; }
            }
#pragma unroll
            for (int m = 1; m <= 8; m <<= 1) {
                float ov = __shfl_xor(v, m, 32);
                int   oc = __shfl_xor(c, m, 32);
                if (ov > v || (ov == v && oc < c)) { v = ov; c = oc; }
            }
            rv[r] = v; rc[r] = c;
        }
        if (l15 == 0) {
#pragma unroll
            for (int r = 0; r < 8; ++r) {
                int row = row0 + r + (half << 3);
                unsigned long long packed =
                    ((unsigned long long)okey(rv[r]) << 32) |
                    (unsigned int)(0xFFFFFFFFu - (unsigned int)rc[r]);
                atomicMax(rowbest + row, packed);
            }
        }
    }

    if (half == 0) {
#pragma unroll
        for (int s = 0; s < 4; ++s) {
            float4 e;
            e.x = cs1[s]; e.y = __int_as_float(ci1[s]); e.z = cs2[s]; e.w = 0.f;
            colpart[(size_t)rg * NPTS + cols[s]] = e;
        }
    }
}

// ---------------------------------------------------------------- merge column partials -> best2 / validity
__global__ __launch_bounds__(256)
void k_colreduce(const float4* __restrict__ colpart,
                 const unsigned long long* __restrict__ rowbest,
                 const int* __restrict__ cnt,
                 int* __restrict__ best2, int* __restrict__ valid) {
    int j = blockIdx.x * 256 + threadIdx.x;
    float s1 = -3.0e38f, s2 = -3.0e38f; int i1 = 0;
    for (int rg = 0; rg < 32; ++rg) {
        float4 e = colpart[(size_t)rg * NPTS + j];
        int idx = __float_as_int(e.y);
        if (e.x > s1) { s2 = fmaxf(s1, e.z); s1 = e.x; i1 = idx; }
        else          { s2 = fmaxf(s2, e.x); }
    }
    unsigned long long rb = rowbest[i1];
    int b1 = (int)(0xFFFFFFFFu - (unsigned int)(rb & 0xFFFFFFFFull));
    float d1 = sqrtf(fmaxf(2.f - 2.f * s1, 0.f));
    float d2 = sqrtf(fmaxf(2.f - 2.f * s2, 0.f));
    float ratio = d1 / d2;
    int ok = (b1 == j) && (ratio <= 0.9f) && (cnt[j] < NPTS / 2);
    best2[j] = i1;
    valid[j] = ok;
}

// ---------------------------------------------------------------- per-neg GEMM: anchors * neg^T row-argmax
// grid (256 row tiles, 6 negs), 256 threads
__global__ __launch_bounds__(256)
void k_gemm_neg(const float* __restrict__ sf, const float* __restrict__ invn,
                const int* __restrict__ best2,
                unsigned long long* __restrict__ negbest) {
    __shared__ float As[16 * LDS_STRIDE];
    const int rt   = blockIdx.x;
    const int neg  = blockIdx.y;
    const int tid  = threadIdx.x;
    const int wave = tid >> 5;
    const int lane = tid & 31;
    const int half = lane >> 4;
    const int l15  = lane & 15;

    const float* Q  = sf;
    const float* NF = sf + (size_t)(2 + neg) * NPTS * DIM;
    const int row0 = rt * 16;

    {   // stage raw anchors q[best2[row]] via async global->LDS
        int r  = tid >> 4;
        int c0 = (tid & 15) * 64;
        int g  = best2[row0 + r];
        const float* src = Q + (size_t)g * DIM + c0;
        float* dst = As + r * LDS_STRIDE + c0;
#pragma unroll
        for (int i = 0; i < 16; ++i)
            async_copy_b128(dst + i * 4, src + i * 4);
        async_wait0();
    }
    __syncthreads();

    // anchor row scales (constant over column iterations)
    float rsc[8];
#pragma unroll
    for (int r = 0; r < 8; ++r) rsc[r] = invn[best2[row0 + r + (half << 3)]];

    float runv[8]; int runc[8];
#pragma unroll
    for (int r = 0; r < 8; ++r) { runv[r] = -3.0e38f; runc[r] = 0; }

    const float* arow = As + l15 * LDS_STRIDE + half * 2;

    for (int it = 0; it < 8; ++it) {
        const int wcol0 = it * 512 + wave * 64;
        float sb[4]; const float* pb[4]; int cols[4];
#pragma unroll
        for (int s = 0; s < 4; ++s) {
            int c   = wcol0 + s * 16 + l15;
            cols[s] = c;
            sb[s]   = invn[(2 + neg) * NPTS + c];
            pb[s]   = NF + (size_t)c * DIM;
        }
        v8f acc[4];
#pragma unroll
        for (int s = 0; s < 4; ++s)
#pragma unroll
            for (int r = 0; r < 8; ++r) acc[s][r] = 0.f;

#pragma unroll 4
        for (int k0 = 0; k0 < DIM; k0 += 4) {
            v2f aF = *(const v2f*)(arow + k0);
#pragma unroll
            for (int s = 0; s < 4; ++s) {
                v2f bF = *(const v2f*)(pb[s] + k0 + half * 2);
                acc[s] = __builtin_amdgcn_wmma_f32_16x16x4_f32(
                    false, aF, false, bF, (short)0, acc[s], false, false);
            }
        }
        // epilogue scales
#pragma unroll
        for (int s = 0; s < 4; ++s)
#pragma unroll
            for (int r = 0; r < 8; ++r) acc[s][r] *= rsc[r] * sb[s];

#pragma unroll
        for (int r = 0; r < 8; ++r) {
            float v = acc[0][r]; int c = cols[0];
#pragma unroll
            for (int s = 1; s < 4; ++s) {
                float w = acc[s][r];
                if (w > v) { v = w; c = cols[s]; }
            }
#pragma unroll
            for (int m = 1; m <= 8; m <<= 1) {
                float ov = __shfl_xor(v, m, 32);
                int   oc = __shfl_xor(c, m, 32);
                if (ov > v || (ov == v && oc < c)) { v = ov; c = oc; }
            }
            if (v > runv[r] || (v == runv[r] && c < runc[r])) { runv[r] = v; runc[r] = c; }
        }
    }
    if (l15 == 0) {
#pragma unroll
        for (int r = 0; r < 8; ++r) {
            int row = row0 + r + (half << 3);
            unsigned long long packed =
                ((unsigned long long)okey(runv[r]) << 32) |
                (unsigned int)(0xFFFFFFFFu - (unsigned int)runc[r]);
            atomicMax(negbest + (size_t)neg * NPTS + row, packed);
        }
    }
}

// ---------------------------------------------------------------- exact per-column hinge loss
__global__ __launch_bounds__(256)
void k_loss_cols(const float* __restrict__ sf, const float* __restrict__ invn,
                 const int* __restrict__ best2, const int* __restrict__ valid,
                 const unsigned long long* __restrict__ negbest,
                 float* __restrict__ colloss) {
    __shared__ float red[256];
    int j   = blockIdx.x;
    int tid = threadIdx.x;
    if (!valid[j]) { if (tid == 0) colloss[j] = 0.f; return; }

    int aidx = best2[j];
    float sa = invn[aidx];
    float sp = invn[NPTS + j];
    float4 av = ((const float4*)(sf + (size_t)aidx * DIM))[tid];
    av.x *= sa; av.y *= sa; av.z *= sa; av.w *= sa;
    float4 pv = ((const float4*)(sf + (size_t)(NPTS + j) * DIM))[tid];
    pv.x *= sp; pv.y *= sp; pv.z *= sp; pv.w *= sp;

    float sums[7];
    {
        float dx = av.x - pv.x, dy = av.y - pv.y, dz = av.z - pv.z, dw = av.w - pv.w;
        sums[0] = dx*dx + dy*dy + dz*dz + dw*dw;
    }
#pragma unroll
    for (int t = 0; t < NNEG; ++t) {
        unsigned long long nb = negbest[(size_t)t * NPTS + j];
        int ni = (int)(0xFFFFFFFFu - (unsigned int)(nb & 0xFFFFFFFFull));
        float sn = invn[(2 + t) * NPTS + ni];
        float4 nv = ((const float4*)(sf + (size_t)(2 + t) * NPTS * DIM + (size_t)ni * DIM))[tid];
        nv.x *= sn; nv.y *= sn; nv.z *= sn; nv.w *= sn;
        float dx = av.x - nv.x, dy = av.y - nv.y, dz = av.z - nv.z, dw = av.w - nv.w;
        sums[1 + t] = dx*dx + dy*dy + dz*dz + dw*dw;
    }

    float res[7];
    for (int s = 0; s < 7; ++s) {
        red[tid] = sums[s];
        __syncthreads();
        for (int off = 128; off >= 1; off >>= 1) {
            if (tid < off) red[tid] += red[tid + off];
            __syncthreads();
        }
        if (tid == 0) res[s] = red[0];
        __syncthreads();
    }
    if (tid == 0) {
        float l = 0.f;
#pragma unroll
        for (int t = 0; t < NNEG; ++t) l += fmaxf(res[0] - res[1 + t] + 1.1f, 0.f);
        colloss[j] = l;
    }
}

// ---------------------------------------------------------------- final deterministic sum
__global__ __launch_bounds__(256)
void k_final(const float* __restrict__ colloss, float* __restrict__ out) {
    __shared__ float red[256];
    int tid = threadIdx.x;
    float s = 0.f;
    for (int i = 0; i < 16; ++i) s += colloss[tid * 16 + i];
    red[tid] = s;
    __syncthreads();
    for (int off = 128; off >= 1; off >>= 1) {
        if (tid < off) red[tid] += red[tid + off];
        __syncthreads();
    }
    if (tid == 0) out[0] = red[0] * 1.0f;   // WEIGHT
}

// ================================================================ launch
extern "C" void kernel_launch(void* const* d_in, const int* in_sizes, int n_in,
                              void* d_out, int out_size, void* d_ws, size_t ws_size,
                              hipStream_t stream) {
    const float* sf  = (const float*)d_in[0];   // [8,4096,1024]
    const float* att = (const float*)d_in[1];   // [8,4096]
    float* out = (float*)d_out;

    char* ws = (char*)d_ws;
    float*              invn    = (float*)ws;                               // 32768 f
    unsigned long long* rowbest = (unsigned long long*)(ws + 131072);       // 4096 u64
    unsigned long long* negbest = rowbest + NPTS;                           // 24576 u64
    float4*             colpart = (float4*)(ws + 360448);                   // 32*4096 f4
    int*                cnt     = (int*)(ws + 360448 + 2097152);            // 4096 i
    int*                best2   = cnt + NPTS;
    int*                valid   = best2 + NPTS;
    float*              colloss = (float*)(valid + NPTS);

    k_invnorm<<<4096, 256, 0, stream>>>(sf, invn);
    k_init<<<(NPTS * (1 + NNEG) + 255) / 256, 256, 0, stream>>>(rowbest, NPTS * (1 + NNEG));
    k_mask<<<16, 256, 0, stream>>>(att + NPTS, cnt);
    k_gemm_qp<<<dim3(8, 32), 256, 0, stream>>>(sf, invn, rowbest, colpart);
    k_colreduce<<<16, 256, 0, stream>>>(colpart, rowbest, cnt, best2, valid);
    k_gemm_neg<<<dim3(256, NNEG), 256, 0, stream>>>(sf, invn, best2, negbest);
    k_loss_cols<<<NPTS, 256, 0, stream>>>(sf, invn, best2, valid, negbest, colloss);
    k_final<<<1, 256, 0, stream>>>(colloss, out);
}